// GemmaAttention_70248485094144
// MI455X (gfx1250) — compile-verified
//
#include <hip/hip_runtime.h>
#include <hip/hip_bf16.h>
#include <math.h>

// ---------------- problem constants ----------------
#define B_      8
#define HID_    3584
#define NH_     16
#define NKV_    8
#define HD_     256
#define MAXSEQ_ 8192
#define QSIZE_  (NH_ * HD_)                // 4096
#define KVSIZE_ (NKV_ * HD_)               // 2048
#define QKVO_   ((NH_ + 2 * NKV_) * HD_)   // 8192
#define SCALE_  0.0625f                    // 256^-0.5
#define EPS_    1e-6f

#define KSPLIT_ 4        // GEMM K-split
#define NSPLIT_ 8        // attention sequence split
#define SPLEN_  (MAXSEQ_ / NSPLIT_)        // 1024
#define LDSTR_  68       // LDS row stride (16B-aligned rows, conflict-free)

typedef __attribute__((ext_vector_type(2))) float v2f;
typedef __attribute__((ext_vector_type(8))) float v8f;

// ---------------- CDNA5 async global->LDS copy ----------------
__device__ __forceinline__ void async_ld_b128(float* lds_ptr, const float* gptr) {
    // low 32 bits of the generic LDS pointer are the DS byte address
    const unsigned int l = (unsigned int)(unsigned long long)lds_ptr;
    asm volatile("global_load_async_to_lds_b128 %0, %1, off"
                 :: "v"(l), "v"(gptr) : "memory");
}
__device__ __forceinline__ void wait_async0() {
    asm volatile("s_wait_asynccnt 0x0" ::: "memory");
}

// =====================================================================
// Kernel 1: D_part[sp][8 x N] = A[8, kbeg:kend] * B[N, kbeg:kend]^T
//   WMMA f32 16x16x4, A padded M=8->16. Double-buffered async-to-LDS
//   staging: chunk i+1 DMA overlaps chunk i WMMA compute.
//   grid = (N/128) * KSPLIT, 256 threads (8 waves x 16 cols).
// =====================================================================
__global__ __launch_bounds__(256)
void gemm8_wmma(const float* __restrict__ Bw,   // [N x K] row-major
                const float* __restrict__ Am,   // [8 x K] row-major
                float* __restrict__ Dpart,      // [KSPLIT][8 x N]
                int N, int klen) {
    __shared__ float lA[2][16 * LDSTR_];
    __shared__ float lB[2][128 * LDSTR_];

    const int t    = threadIdx.x;
    const int wave = t >> 5;
    const int lane = t & 31;
    const int m16  = lane & 15;
    const int half = lane >> 4;
    const int nbx  = N >> 7;
    const int sp   = blockIdx.x / nbx;
    const int nb   = (blockIdx.x % nbx) * 128;
    const int K    = klen * KSPLIT_;
    const int kbeg = sp * klen;
    const int kend = kbeg + klen;

    // zero the A padding rows (8..15) once; never touched by async copies
    for (int i = 8 * LDSTR_ + t; i < 16 * LDSTR_; i += 256) {
        lA[0][i] = 0.f;
        lA[1][i] = 0.f;
    }

    auto stage = [&](int buf, int k0) {
        if (t < 128) {
            const int row = t >> 4;           // 0..7
            const int c4  = (t & 15) * 4;
            async_ld_b128(&lA[buf][row * LDSTR_ + c4],
                          Am + (size_t)row * K + k0 + c4);
        }
        #pragma unroll
        for (int i = 0; i < 8; ++i) {
            const int idx = t + i * 256;      // float4 slots
            const int row = idx >> 4;         // 0..127
            const int c4  = (idx & 15) * 4;
            async_ld_b128(&lB[buf][row * LDSTR_ + c4],
                          Bw + (size_t)(nb + row) * K + k0 + c4);
        }
    };

    v8f acc = {0.f, 0.f, 0.f, 0.f, 0.f, 0.f, 0.f, 0.f};

    stage(0, kbeg);
    wait_async0();
    __syncthreads();          // pad zeros (DScnt) + tile 0 visible to all

    int buf = 0;
    for (int k0 = kbeg; k0 < kend; k0 += 64) {
        // prefetch next chunk into the other buffer while computing
        if (k0 + 64 < kend) stage(buf ^ 1, k0 + 64);

        const float* arow = &lA[buf][m16 * LDSTR_];
        const float* brow = &lB[buf][(wave * 16 + m16) * LDSTR_];
        #pragma unroll
        for (int kk = 0; kk < 64; kk += 4) {
            v2f a, b;
            a.x = arow[kk + 2 * half];
            a.y = arow[kk + 2 * half + 1];
            b.x = brow[kk + 2 * half];
            b.y = brow[kk + 2 * half + 1];
            acc = __builtin_amdgcn_wmma_f32_16x16x4_f32(
                false, a, false, b, (short)0, acc, false, false);
        }

        wait_async0();        // own DMA for next chunk complete
        __syncthreads();      // everyone done reading buf / writing buf^1
        buf ^= 1;
    }

    if (lane < 16) {
        float* D = Dpart + (size_t)sp * 8 * N;
        const int n = nb + wave * 16 + m16;
        #pragma unroll
        for (int v = 0; v < 8; ++v)
            D[(size_t)v * N + n] = acc[v];
    }
}

// =====================================================================
// Kernel 2: sum KSPLIT partials (deterministic; no float atomics)
// =====================================================================
__global__ __launch_bounds__(256)
void reduce_ksplit(const float* __restrict__ part, float* __restrict__ out, int n) {
    const int i = blockIdx.x * 256 + threadIdx.x;
    if (i < n) {
        float s = part[i];
        #pragma unroll
        for (int k = 1; k < KSPLIT_; ++k) s += part[(size_t)k * n + i];
        out[i] = s;
    }
}

// =====================================================================
// Kernel 3: fused RMSNorm + RoPE (+ SCALE for q) per (batch, head)
// =====================================================================
__global__ __launch_bounds__(256)
void normrope(const float* __restrict__ qkv,
              const float* __restrict__ qw,
              const float* __restrict__ kw,
              const float* __restrict__ fcos,
              const float* __restrict__ fsin,
              float* __restrict__ q_ready,    // [B,16,256]
              float* __restrict__ k_ready) {  // [B,8,256]
    __shared__ float red[256];
    __shared__ float sh[256];

    const int b   = blockIdx.x / (NH_ + NKV_);
    const int idx = blockIdx.x % (NH_ + NKV_);
    const int t   = threadIdx.x;

    const float* src; const float* w; float* dst; float scale;
    if (idx < NH_) {
        src   = qkv + (size_t)b * QKVO_ + idx * HD_;
        w     = qw;
        dst   = q_ready + ((size_t)b * NH_ + idx) * HD_;
        scale = SCALE_;
    } else {
        const int kh = idx - NH_;
        src   = qkv + (size_t)b * QKVO_ + QSIZE_ + kh * HD_;
        w     = kw;
        dst   = k_ready + ((size_t)b * NKV_ + kh) * HD_;
        scale = 1.f;
    }

    const float x = src[t];
    red[t] = x * x;
    __syncthreads();
    for (int s = 128; s > 0; s >>= 1) {
        if (t < s) red[t] += red[t + s];
        __syncthreads();
    }
    const float rms = rsqrtf(red[0] * (1.f / 256.f) + EPS_);
    sh[t] = x * rms * (1.f + w[t]);
    __syncthreads();

    float o;
    if (t < 128) o = sh[t] * fcos[t] - sh[t + 128] * fsin[t];
    else         o = sh[t] * fcos[t - 128] + sh[t - 128] * fsin[t - 128];
    dst[t] = o * scale;
}

// =====================================================================
// Kernel 4: flash-decode partial attention.
//   grid = B*NKV*NSPLIT blocks; block owns SPLEN_ positions of one
//   (batch, kv-head) and both Q heads sharing it (K/V streamed once).
//   Row `pos` is patched from fresh k/v (inputs stay read-only).
// =====================================================================
__device__ __forceinline__ float fast_tanh(float x) {
    x = fminf(fmaxf(x, -15.f), 15.f);
    const float e = __expf(2.f * x);
    return (e - 1.f) / (e + 1.f);
}

__global__ __launch_bounds__(256)
void attn_partial(const float* __restrict__ kc,       // [B,8192,8,256]
                  const float* __restrict__ vc,       // [B,8192,8,256]
                  const float* __restrict__ lmask,    // [8192]
                  const float* __restrict__ q_ready,  // [B,16,256] pre-scaled
                  const float* __restrict__ k_ready,  // [B,8,256]
                  const float* __restrict__ qkv,      // [B x 8192] (v_new)
                  const int*   __restrict__ posp,
                  float* __restrict__ pacc,           // [NSPLIT][64][2][256]
                  float* __restrict__ pmax,           // [NSPLIT][128]
                  float* __restrict__ psum) {         // [NSPLIT][128]
    __shared__ float s0[SPLEN_];
    __shared__ float s1[SPLEN_];
    __shared__ float red[512];

    const int sp  = blockIdx.x % NSPLIT_;
    const int bk  = blockIdx.x / NSPLIT_;   // b*8+kvh
    const int kvh = bk & 7;
    const int b   = bk >> 3;
    const int t   = threadIdx.x;
    const int pos = posp[0];
    const int h0  = 2 * kvh;
    const int j0  = sp * SPLEN_;

    // ---------------- pass 1: scores for SPLEN_ positions ----------------
    const int d0 = (t & 7) * 32;
    float q0r[32], q1r[32];
    {
        const float* q0 = q_ready + ((size_t)b * NH_ + h0) * HD_ + d0;
        const float* q1 = q0 + HD_;
        #pragma unroll
        for (int i = 0; i < 32; i += 4) {
            float4 a = *(const float4*)(q0 + i);
            q0r[i] = a.x; q0r[i+1] = a.y; q0r[i+2] = a.z; q0r[i+3] = a.w;
            float4 c = *(const float4*)(q1 + i);
            q1r[i] = c.x; q1r[i+1] = c.y; q1r[i+2] = c.z; q1r[i+3] = c.w;
        }
    }
    const float* kbase = kc + ((size_t)b * MAXSEQ_ * NKV_ + kvh) * HD_;

    for (int jb = 0; jb < SPLEN_; jb += 32) {
        const int jl = jb + (t >> 3);
        const int j  = j0 + jl;
        const float* kr = kbase + (size_t)j * (NKV_ * HD_) + d0;
        float p0 = 0.f, p1 = 0.f;
        #pragma unroll
        for (int i = 0; i < 32; i += 4) {
            float4 kv4 = *(const float4*)(kr + i);
            p0 += kv4.x * q0r[i] + kv4.y * q0r[i+1] + kv4.z * q0r[i+2] + kv4.w * q0r[i+3];
            p1 += kv4.x * q1r[i] + kv4.y * q1r[i+1] + kv4.z * q1r[i+2] + kv4.w * q1r[i+3];
        }
        #pragma unroll
        for (int off = 1; off < 8; off <<= 1) {
            p0 += __shfl_xor(p0, off, 32);
            p1 += __shfl_xor(p1, off, 32);
        }
        if ((t & 7) == 0) {
            const float mk = lmask[j];
            s0[jl] = fast_tanh(p0 * 0.02f) * 50.f + mk;
            s1[jl] = fast_tanh(p1 * 0.02f) * 50.f + mk;
        }
    }
    __syncthreads();

    // patch scores[pos] with the freshly-normed K row (split owning pos)
    if (pos >= j0 && pos < j0 + SPLEN_ && t < 16) {
        const int hh = t >> 3;
        const float* kn = k_ready + ((size_t)b * NKV_ + kvh) * HD_ + (t & 7) * 32;
        const float* qr = q_ready + ((size_t)b * NH_ + h0 + hh) * HD_ + (t & 7) * 32;
        float p = 0.f;
        #pragma unroll
        for (int i = 0; i < 32; ++i) p += kn[i] * qr[i];
        p += __shfl_xor(p, 1, 32);
        p += __shfl_xor(p, 2, 32);
        p += __shfl_xor(p, 4, 32);
        if ((t & 7) == 0) {
            const float v = fast_tanh(p * 0.02f) * 50.f + lmask[pos];
            if (hh == 0) s0[pos - j0] = v; else s1[pos - j0] = v;
        }
    }
    __syncthreads();

    // ---------------- local softmax stats ----------------
    float m0 = -1e30f, m1 = -1e30f;
    for (int i = t; i < SPLEN_; i += 256) {
        m0 = fmaxf(m0, s0[i]);
        m1 = fmaxf(m1, s1[i]);
    }
    red[t] = m0; red[256 + t] = m1;
    __syncthreads();
    for (int s = 128; s > 0; s >>= 1) {
        if (t < s) {
            red[t]       = fmaxf(red[t], red[t + s]);
            red[256 + t] = fmaxf(red[256 + t], red[256 + t + s]);
        }
        __syncthreads();
    }
    m0 = red[0]; m1 = red[256];
    __syncthreads();

    float l0 = 0.f, l1 = 0.f;
    for (int i = t; i < SPLEN_; i += 256) {
        const float e0 = __expf(s0[i] - m0);
        const float e1 = __expf(s1[i] - m1);
        s0[i] = e0; s1[i] = e1;
        l0 += e0; l1 += e1;
    }
    red[t] = l0; red[256 + t] = l1;
    __syncthreads();
    for (int s = 128; s > 0; s >>= 1) {
        if (t < s) {
            red[t]       += red[t + s];
            red[256 + t] += red[256 + t + s];
        }
        __syncthreads();
    }

    // ---------------- pass 2: unnormalized probs x V ----------------
    const float* vbase = vc + ((size_t)b * MAXSEQ_ * NKV_ + kvh) * HD_ + t;
    float a0 = 0.f, a1 = 0.f;
    #pragma unroll 8
    for (int j = 0; j < SPLEN_; ++j) {
        const float v = vbase[(size_t)(j0 + j) * (NKV_ * HD_)];
        a0 += s0[j] * v;
        a1 += s1[j] * v;
    }
    if (pos >= j0 && pos < j0 + SPLEN_) {
        const float vnew = qkv[(size_t)b * QKVO_ + QSIZE_ + KVSIZE_ + kvh * HD_ + t];
        const float vold = vbase[(size_t)pos * (NKV_ * HD_)];
        a0 += s0[pos - j0] * (vnew - vold);
        a1 += s1[pos - j0] * (vnew - vold);
    }

    pacc[(((size_t)sp * 64 + bk) * 2 + 0) * HD_ + t] = a0;
    pacc[(((size_t)sp * 64 + bk) * 2 + 1) * HD_ + t] = a1;
    if (t == 0) {
        pmax[sp * 128 + bk * 2 + 0] = m0;
        pmax[sp * 128 + bk * 2 + 1] = m1;
        psum[sp * 128 + bk * 2 + 0] = red[0];     // l0
        psum[sp * 128 + bk * 2 + 1] = red[256];   // l1
    }
}

// =====================================================================
// Kernel 5: combine the NSPLIT_ partials per head (flash-decode merge)
// =====================================================================
__global__ __launch_bounds__(256)
void attn_combine(const float* __restrict__ pacc,
                  const float* __restrict__ pmax,
                  const float* __restrict__ psum,
                  float* __restrict__ attn_out) {   // [B x 4096]
    const int bk  = blockIdx.x;   // b*8+kvh
    const int b   = bk >> 3;
    const int kvh = bk & 7;
    const int t   = threadIdx.x;

    #pragma unroll
    for (int h = 0; h < 2; ++h) {
        const int hh = bk * 2 + h;
        float M = -1e30f;
        #pragma unroll
        for (int s = 0; s < NSPLIT_; ++s) M = fmaxf(M, pmax[s * 128 + hh]);
        float L = 0.f, a = 0.f;
        #pragma unroll
        for (int s = 0; s < NSPLIT_; ++s) {
            const float w = __expf(pmax[s * 128 + hh] - M);
            L += psum[s * 128 + hh] * w;
            a += pacc[(((size_t)s * 64 + bk) * 2 + h) * HD_ + t] * w;
        }
        attn_out[(size_t)b * QSIZE_ + (2 * kvh + h) * HD_ + t] = a / L;
    }
}

// =====================================================================
// launch
// =====================================================================
extern "C" void kernel_launch(void* const* d_in, const int* in_sizes, int n_in,
                              void* d_out, int out_size, void* d_ws, size_t ws_size,
                              hipStream_t stream) {
    const float* hidden   = (const float*)d_in[0];
    const float* fcos     = (const float*)d_in[1];
    const float* fsin     = (const float*)d_in[2];
    const int*   pos      = (const int*)  d_in[3];
    const float* k_cache  = (const float*)d_in[4];
    const float* v_cache  = (const float*)d_in[5];
    /* d_in[6] mask unused: reference adds local_mask */
    const float* lmask    = (const float*)d_in[7];
    const float* qkv_w    = (const float*)d_in[8];
    const float* o_w      = (const float*)d_in[9];
    const float* q_norm_w = (const float*)d_in[10];
    const float* k_norm_w = (const float*)d_in[11];

    float* ws       = (float*)d_ws;
    float* qkv_part = ws;                                   // 4*8*8192 = 262144
    float* qkv_ws   = qkv_part + (size_t)KSPLIT_ * B_ * QKVO_;      // 65536
    float* q_ready  = qkv_ws   + (size_t)B_ * QKVO_;                // 32768
    float* k_ready  = q_ready  + (size_t)B_ * NH_ * HD_;            // 16384
    float* pacc     = k_ready  + (size_t)B_ * NKV_ * HD_;           // 262144
    float* pmax     = pacc     + (size_t)NSPLIT_ * 64 * 2 * HD_;    // 1024
    float* psum     = pmax     + (size_t)NSPLIT_ * 128;             // 1024
    float* attn_o   = psum     + (size_t)NSPLIT_ * 128;             // 32768
    float* o_part   = attn_o   + (size_t)B_ * QSIZE_;               // 114688

    // 1) QKV projection (split-K WMMA, overlapped async staging) + reduce
    gemm8_wmma<<<(QKVO_ / 128) * KSPLIT_, 256, 0, stream>>>(
        qkv_w, hidden, qkv_part, QKVO_, HID_ / KSPLIT_);
    reduce_ksplit<<<(B_ * QKVO_) / 256, 256, 0, stream>>>(
        qkv_part, qkv_ws, B_ * QKVO_);

    // 2) RMSNorm + RoPE (+ q scaling)
    normrope<<<B_ * (NH_ + NKV_), 256, 0, stream>>>(
        qkv_ws, q_norm_w, k_norm_w, fcos, fsin, q_ready, k_ready);

    // 3) flash-decode attention: 512 partial blocks + merge
    attn_partial<<<B_ * NKV_ * NSPLIT_, 256, 0, stream>>>(
        k_cache, v_cache, lmask, q_ready, k_ready, qkv_ws, pos,
        pacc, pmax, psum);
    attn_combine<<<B_ * NKV_, 256, 0, stream>>>(pacc, pmax, psum, attn_o);

    // 4) output projection (split-K WMMA) + reduce into d_out
    gemm8_wmma<<<(HID_ / 128) * KSPLIT_, 256, 0, stream>>>(
        o_w, attn_o, o_part, HID_, QSIZE_ / KSPLIT_);
    reduce_ksplit<<<(B_ * HID_) / 256, 256, 0, stream>>>(
        o_part, (float*)d_out, B_ * HID_);
}